// MambaBlock_84473416778379
// MI455X (gfx1250) — compile-verified
//
#include <hip/hip_runtime.h>
#include <hip/hip_bf16.h>
#include <math.h>

typedef __attribute__((ext_vector_type(2))) float v2f;
typedef __attribute__((ext_vector_type(4))) float v4f;
typedef __attribute__((ext_vector_type(8))) float v8f;

#define D_MODEL 2048
#define D_STATE 16
#define D_INNER 4096
#define D_CONV  4
#define DT_RANK 128
#define BATCH   256
#define EPS     1e-5f

// ---------------------------------------------------------------------------
// 1) RMSNorm: one block per row (2048 cols)
// ---------------------------------------------------------------------------
__global__ void rmsnorm_kernel(const float* __restrict__ x,
                               const float* __restrict__ w,
                               float* __restrict__ xn) {
    __shared__ float red[8];
    const int row = blockIdx.x;
    const float* xr = x + (size_t)row * D_MODEL;
    float ss = 0.f;
    for (int i = threadIdx.x; i < D_MODEL; i += blockDim.x) {
        float v = xr[i];
        ss += v * v;
    }
    for (int off = 16; off > 0; off >>= 1) ss += __shfl_down(ss, off, 32);
    const int wave = threadIdx.x >> 5, lane = threadIdx.x & 31;
    if (lane == 0) red[wave] = ss;
    __syncthreads();
    if (threadIdx.x == 0) {
        float t = 0.f;
        for (int i = 0; i < 8; ++i) t += red[i];
        red[0] = rsqrtf(t * (1.0f / D_MODEL) + EPS);
    }
    __syncthreads();
    const float scale = red[0];
    float* out = xn + (size_t)row * D_MODEL;
    for (int i = threadIdx.x; i < D_MODEL; i += blockDim.x)
        out[i] = xr[i] * scale * w[i];
}

// ---------------------------------------------------------------------------
// 2) fp32 WMMA GEMM:  C[M,N] = A[M,K(lda)] * W[N,K(ldw)]^T
//    One 16 x (16*NT) C strip per wave. K stepped by 8 using b128 loads:
//    per lane, A/B fragments come from the SAME K offsets, so splitting a
//    v4f into {xy}/{zw} across two WMMAs covers the full 8-K range (summation
//    reordered only). act: 0 = none, 1 = softplus(v + bias[n]).
// ---------------------------------------------------------------------------
template <int NT>
__global__ void gemm_wmma_f32(const float* __restrict__ A, int lda,
                              const float* __restrict__ W, int ldw,
                              float* __restrict__ C, int ldc,
                              int M, int N, int K,
                              const float* __restrict__ bias, int act) {
    const int wave = threadIdx.x >> 5;
    const int lane = threadIdx.x & 31;
    const int groups_n = N / (16 * NT);
    const int total = (M >> 4) * groups_n;
    const int tile = blockIdx.x * (blockDim.x >> 5) + wave;
    if (tile >= total) return;                 // wave-uniform: EXEC stays all-1s
    const int m0 = (tile / groups_n) << 4;
    const int n0 = (tile % groups_n) * (16 * NT);
    const int half = lane >> 4;
    const int l    = lane & 15;

    // half=0 lanes carry K = {k0..k0+3}; half=1 lanes carry K = {k0+4..k0+7}
    const float* arow  = A + (size_t)(m0 + l) * lda + 4 * half;
    const float* wbase = W + (size_t)(n0 + l) * ldw + 4 * half;

    v8f acc[NT];
    #pragma unroll
    for (int t = 0; t < NT; ++t) acc[t] = (v8f){0.f,0.f,0.f,0.f,0.f,0.f,0.f,0.f};

    for (int k0 = 0; k0 < K; k0 += 8) {
        v4f av = *(const v4f*)(arow + k0);
        v2f a01 = {av.x, av.y};
        v2f a23 = {av.z, av.w};
        #pragma unroll
        for (int t = 0; t < NT; ++t) {
            const float* wr = wbase + (size_t)t * 16 * ldw + k0;
            v4f bv = *(const v4f*)wr;
            __builtin_prefetch(wr + 64, 0, 1);    // weight stream, ~8 blocks ahead
            v2f b01 = {bv.x, bv.y};
            v2f b23 = {bv.z, bv.w};
            acc[t] = __builtin_amdgcn_wmma_f32_16x16x4_f32(
                         false, a01, false, b01, (short)0, acc[t], false, false);
            acc[t] = __builtin_amdgcn_wmma_f32_16x16x4_f32(
                         false, a23, false, b23, (short)0, acc[t], false, false);
        }
    }

    // C/D layout: VGPR r, lanes 0-15 -> row m0+r; lanes 16-31 -> row m0+8+r
    #pragma unroll
    for (int t = 0; t < NT; ++t) {
        const int col = n0 + 16 * t + l;
        float* crow = C + (size_t)(m0 + 8 * half) * ldc + col;
        const float bv = (act == 1) ? bias[col] : 0.f;
        #pragma unroll
        for (int r = 0; r < 8; ++r) {
            float v = acc[t][r];
            if (act == 1) {
                v += bv;
                v = (v > 20.f) ? v : log1pf(expf(v));   // softplus
            }
            crow[(size_t)r * ldc] = v;
        }
    }
}

// ---------------------------------------------------------------------------
// 3) Depthwise conv shift + bias + SiLU. Writes conv_state_new into d_out.
// ---------------------------------------------------------------------------
__global__ void conv_silu_kernel(const float* __restrict__ conv_state,
                                 const float* __restrict__ xz,     // [B, 2*D_INNER]
                                 const float* __restrict__ conv_w,
                                 const float* __restrict__ conv_b,
                                 float* __restrict__ conv_state_new,
                                 float* __restrict__ x_conv) {
    const int idx = blockIdx.x * blockDim.x + threadIdx.x;   // b*D_INNER + c
    if (idx >= BATCH * D_INNER) return;
    const int b = idx / D_INNER;
    const int c = idx - b * D_INNER;
    const float* cs = conv_state + (size_t)idx * D_CONV;
    const float s0 = cs[1], s1 = cs[2], s2 = cs[3];
    const float xv = xz[(size_t)b * (2 * D_INNER) + c];
    const float* wc = conv_w + (size_t)c * D_CONV;
    float acc = s0 * wc[0] + s1 * wc[1] + s2 * wc[2] + xv * wc[3] + conv_b[c];
    float* cn = conv_state_new + (size_t)idx * D_CONV;
    cn[0] = s0; cn[1] = s1; cn[2] = s2; cn[3] = xv;
    x_conv[idx] = acc / (1.f + expf(-acc));                   // silu
}

// ---------------------------------------------------------------------------
// 4) SSM recurrence + output dot + D-skip + SiLU(z) gate.
// ---------------------------------------------------------------------------
__global__ void ssm_kernel(const float* __restrict__ ssm_state,
                           const float* __restrict__ delta,
                           const float* __restrict__ x_conv,
                           const float* __restrict__ dt_bc,     // [B, 160]
                           const float* __restrict__ A_log,
                           const float* __restrict__ Dp,
                           const float* __restrict__ xz,        // z = xz[:, D_INNER:]
                           float* __restrict__ ssm_state_new,
                           float* __restrict__ y) {
    const int idx = blockIdx.x * blockDim.x + threadIdx.x;     // b*D_INNER + c
    if (idx >= BATCH * D_INNER) return;
    const int b = idx / D_INNER;
    const int c = idx - b * D_INNER;
    const float d  = delta[idx];
    const float xc = x_conv[idx];
    const float* Bv = dt_bc + (size_t)b * (DT_RANK + 2 * D_STATE) + DT_RANK;
    const float* Cv = Bv + D_STATE;
    const float* st = ssm_state + (size_t)idx * D_STATE;
    float* stn      = ssm_state_new + (size_t)idx * D_STATE;
    const float* al = A_log + (size_t)c * D_STATE;
    float acc = 0.f;
    #pragma unroll
    for (int s = 0; s < D_STATE; ++s) {
        const float Aval  = -expf(al[s]);
        const float a_bar = expf(d * Aval);
        const float v = a_bar * st[s] + d * Bv[s] * xc;
        stn[s] = v;
        acc += v * Cv[s];
    }
    acc += Dp[c] * xc;
    const float z  = xz[(size_t)b * (2 * D_INNER) + D_INNER + c];
    const float sz = z / (1.f + expf(-z));
    y[idx] = acc * sz;
}

// ---------------------------------------------------------------------------
// Launcher
// ---------------------------------------------------------------------------
extern "C" void kernel_launch(void* const* d_in, const int* in_sizes, int n_in,
                              void* d_out, int out_size, void* d_ws, size_t ws_size,
                              hipStream_t stream) {
    const float* x_t        = (const float*)d_in[0];
    const float* conv_state = (const float*)d_in[1];
    const float* ssm_state  = (const float*)d_in[2];
    const float* norm_w     = (const float*)d_in[3];
    const float* in_proj_w  = (const float*)d_in[4];
    const float* out_proj_w = (const float*)d_in[5];
    const float* conv_w     = (const float*)d_in[6];
    const float* conv_b     = (const float*)d_in[7];
    const float* x_proj_w   = (const float*)d_in[8];
    const float* dt_proj_w  = (const float*)d_in[9];
    const float* dt_proj_b  = (const float*)d_in[10];
    const float* A_log      = (const float*)d_in[11];
    const float* Dp         = (const float*)d_in[12];

    float* out = (float*)d_out;                                // [256, 2048]
    float* conv_state_new = out + (size_t)BATCH * D_MODEL;     // [256, 4096, 4]
    float* ssm_state_new  = conv_state_new + (size_t)BATCH * D_INNER * D_CONV;

    float* ws     = (float*)d_ws;
    float* xn     = ws;
    float* xz     = xn + (size_t)BATCH * D_MODEL;
    float* x_conv = xz + (size_t)BATCH * 2 * D_INNER;
    float* dt_bc  = x_conv + (size_t)BATCH * D_INNER;
    float* delta  = dt_bc + (size_t)BATCH * (DT_RANK + 2 * D_STATE);
    float* y      = delta + (size_t)BATCH * D_INNER;

    const int BLK = 256;                // 8 waves per block
    const int WPB = BLK / 32;

    // 1) RMSNorm
    rmsnorm_kernel<<<BATCH, BLK, 0, stream>>>(x_t, norm_w, xn);

    // 2) xz = xn @ in_proj_w^T   (M=256, N=8192, K=2048), NT=4
    {
        int tiles = (BATCH / 16) * ((2 * D_INNER) / 64);
        gemm_wmma_f32<4><<<(tiles + WPB - 1) / WPB, BLK, 0, stream>>>(
            xn, D_MODEL, in_proj_w, D_MODEL, xz, 2 * D_INNER,
            BATCH, 2 * D_INNER, D_MODEL, nullptr, 0);
    }

    // 3) conv shift + silu  (writes conv_state_new to d_out)
    conv_silu_kernel<<<(BATCH * D_INNER + BLK - 1) / BLK, BLK, 0, stream>>>(
        conv_state, xz, conv_w, conv_b, conv_state_new, x_conv);

    // 4) dt_bc = x_conv @ x_proj_w^T  (M=256, N=160, K=4096), NT=2
    {
        int N = DT_RANK + 2 * D_STATE;  // 160
        int tiles = (BATCH / 16) * (N / 32);
        gemm_wmma_f32<2><<<(tiles + WPB - 1) / WPB, BLK, 0, stream>>>(
            x_conv, D_INNER, x_proj_w, D_INNER, dt_bc, N,
            BATCH, N, D_INNER, nullptr, 0);
    }

    // 5) delta = softplus(dt @ dt_proj_w^T + b)  (M=256, N=4096, K=128, lda=160), NT=4
    {
        int tiles = (BATCH / 16) * (D_INNER / 64);
        gemm_wmma_f32<4><<<(tiles + WPB - 1) / WPB, BLK, 0, stream>>>(
            dt_bc, DT_RANK + 2 * D_STATE, dt_proj_w, DT_RANK, delta, D_INNER,
            BATCH, D_INNER, DT_RANK, dt_proj_b, 1);
    }

    // 6) SSM recurrence + gate  (writes ssm_state_new to d_out)
    ssm_kernel<<<(BATCH * D_INNER + BLK - 1) / BLK, BLK, 0, stream>>>(
        ssm_state, delta, x_conv, dt_bc, A_log, Dp, xz, ssm_state_new, y);

    // 7) out = y @ out_proj_w^T  (M=256, N=2048, K=4096), NT=4
    {
        int tiles = (BATCH / 16) * (D_MODEL / 64);
        gemm_wmma_f32<4><<<(tiles + WPB - 1) / WPB, BLK, 0, stream>>>(
            y, D_INNER, out_proj_w, D_INNER, out, D_MODEL,
            BATCH, D_MODEL, D_INNER, nullptr, 0);
    }
}